// LearnableSTFTFunction_84559316124469
// MI455X (gfx1250) — compile-verified
//
#include <hip/hip_runtime.h>

typedef __attribute__((ext_vector_type(2))) float v2f;
typedef __attribute__((ext_vector_type(8))) float v8f;

#define B_   2
#define T_   2000
#define K_   200
#define L_   101
#define TROW (K_ * L_)   // 20200 floats per t-row of output

// out[b, t, m, l] = x_pad[b, t+m] * W[l, m]
// One WMMA_F32_16X16X4_F32 per 16(t) x 16(l) tile, rank-1 (only K=0 live):
//   A[i][0] = x[b, t0+i+m]   (lanes 0-15, VGPR0)
//   B[0][j] = W[l0+j, m]     (lanes 0-15, VGPR0)
// Grid: b(2) * t_tiles(125) * m_groups(25) = 6250 blocks, 256 threads (8 waves),
// each wave owns one m; 7 l-tiles of 16 cover L=101 (last tile store-masked).
__global__ __launch_bounds__(256) void stft_outer_wmma(
    const float* __restrict__ x, const float* __restrict__ W,
    float* __restrict__ out)
{
    __shared__ float smem[8 * L_];   // Wt[c][l] = W[l][m0+c], 3232 B

    const int tid   = threadIdx.x;
    const int wave  = tid >> 5;
    const int lane  = tid & 31;
    const int hlane = lane & 15;

    int bid = blockIdx.x;
    const int mg = bid % 25;  bid /= 25;   // m group (8 m's)
    const int tb = bid % 125; bid /= 125;  // t tile (16 t's)
    const int b  = bid;                    // batch
    const int t0 = tb * 16;
    const int m0 = mg * 8;

    // Stage the 8 needed columns of W into LDS, transposed: smem[c*L_ + l].
    // Global reads: 8 consecutive floats per l (32B segments, L2-resident W).
    // LDS writes: dword stride 101 -> odd mod 64 banks -> conflict-free.
    for (int idx = tid; idx < 8 * L_; idx += 256) {
        const int c = idx & 7;
        const int l = idx >> 3;
        smem[c * L_ + l] = W[l * K_ + m0 + c];
    }
    __syncthreads();

    const int m = m0 + wave;

    // A operand: only (M=lane, K=0) live in lanes 0-15; zero-pad handles t+m >= T.
    v2f a; a.x = 0.0f; a.y = 0.0f;
    const int tt = t0 + hlane + m;
    a.x = (lane < 16 && tt < T_) ? x[b * T_ + tt] : 0.0f;

    // Element index of out[b, t0, m, 0]; row i adds i*TROW, col adds l.
    const size_t rowbase = ((size_t)(b * T_ + t0) * K_ + m) * (size_t)L_;

    #pragma unroll
    for (int lt = 0; lt < 7; ++lt) {
        const int l0 = lt * 16;
        const int l  = l0 + hlane;

        // B operand: only (K=0, N=lane) live in lanes 0-15.
        v2f bv; bv.x = 0.0f; bv.y = 0.0f;
        const int   lc = (l < L_) ? l : (L_ - 1);
        const float wv = smem[wave * L_ + lc];
        bv.x = (lane < 16 && l < L_) ? wv : 0.0f;

        v8f c = {};
        // D[i][j] = x[b, t0+i+m] * W[l0+j, m]  (16x16 tile)
        v8f d = __builtin_amdgcn_wmma_f32_16x16x4_f32(
            /*neg_a=*/false, a, /*neg_b=*/false, bv,
            /*c_mod=*/(short)0, c, /*reuse_a=*/false, /*reuse_b=*/false);

        // D layout: VGPR r = rows {r, r+8}; lanes 0-15 -> row r, cols 0-15;
        // lanes 16-31 -> row r+8, cols 0-15. Streaming NT stores: two 64 B
        // coalesced segments per store instruction.
        if (l < L_) {
            float* p = out + rowbase
                           + (size_t)((lane >> 4) * 8) * TROW
                           + (size_t)(l0 + hlane);
            #pragma unroll
            for (int r = 0; r < 8; ++r)
                __builtin_nontemporal_store(d[r], p + (size_t)r * TROW);
        }
    }
}

extern "C" void kernel_launch(void* const* d_in, const int* in_sizes, int n_in,
                              void* d_out, int out_size, void* d_ws, size_t ws_size,
                              hipStream_t stream)
{
    (void)in_sizes; (void)n_in; (void)out_size; (void)d_ws; (void)ws_size;
    const float* x = (const float*)d_in[0];   // (2, 1, 2000) fp32
    const float* W = (const float*)d_in[1];   // (101, 200)   fp32
    float* out = (float*)d_out;               // (2,1,2000,200,101) fp32

    const int grid = B_ * (T_ / 16) * (K_ / 8);   // 6250
    stft_outer_wmma<<<grid, 256, 0, stream>>>(x, W, out);
}